// PoseAttention_84524956385434
// MI455X (gfx1250) — compile-verified
//
#include <hip/hip_runtime.h>

typedef float v2f __attribute__((ext_vector_type(2)));
typedef float v8f __attribute__((ext_vector_type(8)));
typedef unsigned int v4u __attribute__((ext_vector_type(4)));
typedef int v4i __attribute__((ext_vector_type(4)));
typedef int v8i __attribute__((ext_vector_type(8)));

__device__ __forceinline__ v8f wmma4(v2f a, v2f b, v8f c) {
    // D = A(16x4 f32) * B(4x16 f32) + C(16x16 f32)
    return __builtin_amdgcn_wmma_f32_16x16x4_f32(
        false, a, false, b, (short)0, c, false, false);
}

#define HW 65536      // 256*256
#define DM 256
#define CIN 512

#if defined(__HIP_DEVICE_COMPILE__) && __has_builtin(__builtin_amdgcn_tensor_load_to_lds) && __has_builtin(__builtin_amdgcn_s_wait_tensorcnt)
#define HAVE_TDM 1
#else
#define HAVE_TDM 0
#endif

#if HAVE_TDM
// Issue one 2D TDM tile load: tile1 rows x tile0 elems (4B), row stride in elems.
// flags_w0 packs D# group1 word0: data_size=4B (2<<16) | pad fields.
__device__ __forceinline__ void tdm_load_2d(unsigned lds_byte, const void* gp,
                                            unsigned tile0, unsigned tile1,
                                            unsigned long long stride0,
                                            unsigned flags_w0)
{
    unsigned long long ga = (unsigned long long)(uintptr_t)gp;
    v4u g0;
    g0[0] = 1u;                                   // count=1 (valid descriptor)
    g0[1] = lds_byte;                             // lds_addr
    g0[2] = (unsigned)ga;                         // global_addr[31:0]
    g0[3] = (unsigned)((ga >> 32) & 0x1FFFFFFu) | 0x80000000u; // addr[56:32]|type=2
    const unsigned td0 = 0x0FFFFFFFu, td1 = 0x0FFFFFFFu;       // no OOB clipping
    v8i g1;
    g1[0] = (int)flags_w0;                              // wg_mask=0|data_size|pad
    g1[1] = (int)((td0 & 0xFFFFu) << 16);               // tensor_dim0[15:0]
    g1[2] = (int)((td0 >> 16) | ((td1 & 0xFFFFu) << 16));
    g1[3] = (int)((td1 >> 16) | (tile0 << 16));         // tile_dim0
    g1[4] = (int)(tile1 & 0xFFFFu);                     // tile_dim1 (tile_dim2=0)
    g1[5] = (int)(unsigned)(stride0 & 0xFFFFFFFFull);   // tensor_dim0_stride lo
    g1[6] = (int)(unsigned)((stride0 >> 32) & 0xFFFFull);
    g1[7] = 0;
    v4i gz4 = {0, 0, 0, 0};
    v8i gz8 = {0, 0, 0, 0, 0, 0, 0, 0};
    // clang-23 / therock-10.0 form: 6 args (g0, g1, g2, g3, extra, cpol)
    __builtin_amdgcn_tensor_load_to_lds(g0, g1, gz4, gz4, gz8, 0);
}
#endif

// ---------------------------------------------------------------------------
// Kernel 0a: precompute M = scale * Wq^T Wk  (packed K-pairs)  and
//            W2  = fc_w @ wv_w               (packed K-pairs, transposed)
// ---------------------------------------------------------------------------
__global__ __launch_bounds__(256) void precompute_mats(
    const float* __restrict__ wq, const float* __restrict__ wk,
    const float* __restrict__ wv, const float* __restrict__ fc,
    float* __restrict__ Mp, float* __restrict__ W2p)
{
    const int row = blockIdx.x;    // K-dim index (a for M, c for W2)
    const int col = threadIdx.x;   // N-dim index
    if (blockIdx.y == 0) {
        float s = 0.f;
        for (int j = 0; j < DM; ++j) s += wq[j * DM + row] * wk[j * DM + col];
        Mp[((row >> 1) * DM + col) * 2 + (row & 1)] = s * 0.125f;
    } else {
        float s = 0.f;
        for (int j = 0; j < DM; ++j) s += fc[col * DM + j] * wv[j * DM + row];
        W2p[((row >> 1) * DM + col) * 2 + (row & 1)] = s;   // row=c, col=o
    }
}

// ---------------------------------------------------------------------------
// Kernel 0b: cvec = scale * Wk^T bq ; c2 = fc_w@wv_b + fc_b ; sconst = scale*bq.bk
// ---------------------------------------------------------------------------
__global__ __launch_bounds__(256) void precompute_vecs(
    const float* __restrict__ wk, const float* __restrict__ wqb,
    const float* __restrict__ wkb, const float* __restrict__ fc,
    const float* __restrict__ wvb, const float* __restrict__ fcb,
    float* __restrict__ cvec, float* __restrict__ c2, float* __restrict__ scst)
{
    const int t = threadIdx.x;
    float s = 0.f;
    for (int j = 0; j < DM; ++j) s += wqb[j] * wk[j * DM + t];
    cvec[t] = s * 0.125f;
    float s2 = 0.f;
    for (int j = 0; j < DM; ++j) s2 += fc[t * DM + j] * wvb[j];
    c2[t] = s2 + fcb[t];
    if (t == 0) {
        float d = 0.f;
        for (int j = 0; j < DM; ++j) d += wqb[j] * wkb[j];
        *scst = d * 0.125f;
    }
}

// ---------------------------------------------------------------------------
// Kernel 1: 1x1 conv (512->256) + 2D positional encoding.
// Grid: 2048 = B(2)*H(256)*W/64(4). Block 256 (8 waves).
// TDM double-buffered staging of feature tile [32c x 64w] and weight tile
// [256dch x 32c] (padded to 34-float rows by TDM). WMMA f32 16x16x4 core.
// Output layout: x[b][hc][wc][q][p][dch].
// ---------------------------------------------------------------------------
__global__ __launch_bounds__(256) void conv_pe(
    const float* __restrict__ in, const float* __restrict__ cw,
    const float* __restrict__ cb, float* __restrict__ xout)
{
    __shared__ __align__(16) float ft[2][32][64];    // [c-chunk][w-local]
    __shared__ __align__(16) float wl[2][256][34];   // [dch][c-chunk] pad->34

    const int t = threadIdx.x;
    const int lane = t & 31, wv = t >> 5;
    const int l15 = lane & 15, kh = lane >> 4;

    const int wg = blockIdx.x;
    const int b  = wg >> 10;
    const int h  = (wg >> 2) & 255;
    const int w0 = (wg & 3) << 6;

    const float* fin = in + (size_t)b * CIN * HW + (size_t)h * 256 + w0;

    const v8f vz = {0.f,0.f,0.f,0.f,0.f,0.f,0.f,0.f};
    v8f acc[4][2];
    #pragma unroll
    for (int m = 0; m < 4; ++m)
        #pragma unroll
        for (int n = 0; n < 2; ++n) acc[m][n] = vz;

    // stage one 32-cin chunk into buffer bufi
    auto stage = [&](int bufi, int kc) {
#if HAVE_TDM
        if (t < 32) {   // one wave issues the DMA; EXEC-independent
            tdm_load_2d((unsigned)(uintptr_t)&ft[bufi][0][0],
                        fin + (size_t)kc * HW, 64u, 32u, 65536ull, 0x00020000u);
            tdm_load_2d((unsigned)(uintptr_t)&wl[bufi][0][0],
                        cw + kc, 32u, 256u, 512ull,
                        0x00020000u | (1u << 20) | (4u << 22) | (1u << 25));
        }
#else
        for (int i = t; i < 32 * 64; i += 256) {
            int r = i >> 6, cc = i & 63;
            ft[bufi][r][cc] = fin[(size_t)(kc + r) * HW + cc];
        }
        for (int i = t; i < 256 * 32; i += 256) {
            int dch = i >> 5, cc = i & 31;
            wl[bufi][dch][cc] = cw[(size_t)dch * CIN + (kc + cc)];
        }
#endif
    };

    stage(0, 0);
    int cur = 0;
    for (int kc = 0; kc < CIN; kc += 32, cur ^= 1) {
        const bool more = (kc + 32) < CIN;
        if (more) {
            stage(cur ^ 1, kc + 32);
            __builtin_prefetch((const void*)(fin + (size_t)(kc + 64) * HW), 0, 1);
        }
#if HAVE_TDM
        if (more) __builtin_amdgcn_s_wait_tensorcnt(2);   // current chunk done
        else      __builtin_amdgcn_s_wait_tensorcnt(0);
#endif
        __syncthreads();   // current chunk visible to all waves

        #pragma unroll
        for (int ks = 0; ks < 32; ks += 4) {
            v2f afr[4], bfr[2];
            #pragma unroll
            for (int m = 0; m < 4; ++m) {
                afr[m].x = ft[cur][ks + 2 * kh][m * 16 + l15];
                afr[m].y = ft[cur][ks + 2 * kh + 1][m * 16 + l15];
            }
            #pragma unroll
            for (int n = 0; n < 2; ++n) {
                const int dch = (wv * 2 + n) * 16 + l15;
                bfr[n] = *(const v2f*)&wl[cur][dch][ks + 2 * kh];
            }
            #pragma unroll
            for (int m = 0; m < 4; ++m)
                #pragma unroll
                for (int n = 0; n < 2; ++n)
                    acc[m][n] = wmma4(afr[m], bfr[n], acc[m][n]);
        }
        __syncthreads();   // buffer 'cur' free for reuse
    }

    // epilogue: + bias + positional encoding, store to chunked layout
    const int hc = h >> 6, p = h & 63, wc = w0 >> 6;
    float* xc = xout + (size_t)(b * 16 + hc * 4 + wc) * 4096 * DM + (size_t)p * DM;

    #pragma unroll
    for (int n = 0; n < 2; ++n) {
        const int dch = (wv * 2 + n) * 16 + l15;
        const float bias = cb[dch];
        const bool lo = dch < 128;
        const bool ev = (dch & 1) == 0;
        float div = 0.f, pes = 0.f;
        if (lo) {
            div = __expf(-(float)(dch & ~1) * 0.035977892f); // ln(1e4)/256
            if (ev) pes = __sinf((float)h * div);
        }
        #pragma unroll
        for (int m = 0; m < 4; ++m) {
            #pragma unroll
            for (int r = 0; r < 8; ++r) {
                const int sp = m * 16 + r + kh * 8;            // q = w-local
                float pe = 0.f;
                if (lo) pe = ev ? pes : __cosf((float)(w0 + sp) * div);
                xc[(size_t)sp * 64 * DM + dch] = acc[m][n][r] + bias + pe;
            }
        }
    }
}

// ---------------------------------------------------------------------------
// Kernel 2: fused attention per (chunk, q). Grid 2048, block 256 (8 waves).
//   T = X*M ; S = T*X^T + g ; softmax(S) ; O = S*X ; out = O*W2^T + c2
// X slab loaded by TDM with on-the-fly padding (4 DW per 256 DW -> 260 stride).
// ---------------------------------------------------------------------------
#define XS 260
__global__ __launch_bounds__(256) void attn_fused(
    const float* __restrict__ x, const float* __restrict__ Mp,
    const float* __restrict__ W2p, const float* __restrict__ cvec,
    const float* __restrict__ c2, const float* __restrict__ scst,
    float* __restrict__ out)
{
    __shared__ __align__(16) float X[64][XS];   // tokens x d (later final out)
    __shared__ __align__(16) float T[64][XS];   // X*M, later A*X
    __shared__ __align__(16) float S[64][68];   // scores / attention
    __shared__ __align__(16) float gv[64];

    const int t = threadIdx.x, lane = t & 31, wv = t >> 5;
    const int l15 = lane & 15, kh = lane >> 4;
    const int wg = blockIdx.x;
    const int q = wg & 63, chunk = wg >> 6;
    const int b = chunk >> 4, hc = (chunk >> 2) & 3, wc = chunk & 3;

    const v8f vz = {0.f,0.f,0.f,0.f,0.f,0.f,0.f,0.f};

    // ---- load X: 64 tokens x 256 d (contiguous 64 KB in memory) ----
#if HAVE_TDM
    if (t < 32)
        tdm_load_2d((unsigned)(uintptr_t)&X[0][0],
                    x + (size_t)(chunk * 4096 + q * 64) * DM,
                    16384u, 1u, 16384ull,
                    0x00020000u | (1u << 20) | (7u << 22) | (3u << 25));
    __builtin_amdgcn_s_wait_tensorcnt(0);
#else
    const float4* xs = (const float4*)(x + (size_t)(chunk * 4096 + q * 64) * DM);
    for (int i = t; i < 4096; i += 256) {
        int p = i >> 6, c4 = i & 63;
        float4 v = xs[p * 64 + c4];
        *(float4*)&X[p][c4 * 4] = v;
    }
#endif
    __syncthreads();

    // ---- g(P) = scale*(x_P . Wk^T bq) + scale*bq.bk ----
    if (t < 64) {
        float s = *scst;
        for (int c = 0; c < DM; ++c) s += X[t][c] * cvec[c];
        gv[t] = s;
    }

    // ---- phase 1: T = X * M  (M=64 p, N=256, K=256) ----
    {
        v8f acc[4][2];
        #pragma unroll
        for (int m = 0; m < 4; ++m)
            #pragma unroll
            for (int n = 0; n < 2; ++n) acc[m][n] = vz;
        for (int k = 0; k < DM; k += 4) {
            v2f afr[4], bfr[2];
            #pragma unroll
            for (int m = 0; m < 4; ++m)
                afr[m] = *(const v2f*)&X[m * 16 + l15][k + 2 * kh];
            #pragma unroll
            for (int n = 0; n < 2; ++n) {
                const int col = (wv * 2 + n) * 16 + l15;
                bfr[n] = *(const v2f*)&Mp[(size_t)((k >> 1) + kh) * 512 + col * 2];
            }
            #pragma unroll
            for (int m = 0; m < 4; ++m)
                #pragma unroll
                for (int n = 0; n < 2; ++n)
                    acc[m][n] = wmma4(afr[m], bfr[n], acc[m][n]);
        }
        #pragma unroll
        for (int n = 0; n < 2; ++n) {
            const int col = (wv * 2 + n) * 16 + l15;
            #pragma unroll
            for (int m = 0; m < 4; ++m)
                #pragma unroll
                for (int r = 0; r < 8; ++r)
                    T[m * 16 + r + kh * 8][col] = acc[m][n][r];
        }
    }
    __syncthreads();

    // ---- phase 2: S = T * X^T + g  (64x64, K=256); 2 tiles/wave ----
    {
        v8f sc2[2]; sc2[0] = vz; sc2[1] = vz;
        const int mt = (wv * 2) >> 2;              // same for both tiles
        const int nt0 = (wv * 2) & 3, nt1 = (wv * 2 + 1) & 3;
        for (int k = 0; k < DM; k += 4) {
            v2f a0 = *(const v2f*)&T[mt  * 16 + l15][k + 2 * kh];
            v2f b0 = *(const v2f*)&X[nt0 * 16 + l15][k + 2 * kh];
            sc2[0] = wmma4(a0, b0, sc2[0]);
            v2f b1 = *(const v2f*)&X[nt1 * 16 + l15][k + 2 * kh];
            sc2[1] = wmma4(a0, b1, sc2[1]);
        }
        #pragma unroll
        for (int s = 0; s < 2; ++s) {
            const int nt = (wv * 2 + s) & 3;
            const int P = nt * 16 + l15;
            const float g = gv[P];
            #pragma unroll
            for (int r = 0; r < 8; ++r)
                S[mt * 16 + r + kh * 8][P] = sc2[s][r] + g;
        }
    }
    __syncthreads();

    // ---- softmax over P (rows of S) ----
    if (t < 64) {
        float mx = -1e30f;
        for (int j = 0; j < 64; ++j) mx = fmaxf(mx, S[t][j]);
        float sum = 0.f;
        for (int j = 0; j < 64; ++j) {
            float e = __expf(S[t][j] - mx);
            S[t][j] = e; sum += e;
        }
        const float inv = 1.0f / sum;
        for (int j = 0; j < 64; ++j) S[t][j] *= inv;
    }
    __syncthreads();

    // ---- phase 3: O = A * X  (M=64 p, N=256 d, K=64) -> T ----
    {
        v8f acc[4][2];
        #pragma unroll
        for (int m = 0; m < 4; ++m)
            #pragma unroll
            for (int n = 0; n < 2; ++n) acc[m][n] = vz;
        for (int k = 0; k < 64; k += 4) {
            v2f afr[4], bfr[2];
            #pragma unroll
            for (int m = 0; m < 4; ++m)
                afr[m] = *(const v2f*)&S[m * 16 + l15][k + 2 * kh];
            #pragma unroll
            for (int n = 0; n < 2; ++n) {
                const int col = (wv * 2 + n) * 16 + l15;
                bfr[n].x = X[k + 2 * kh][col];
                bfr[n].y = X[k + 2 * kh + 1][col];
            }
            #pragma unroll
            for (int m = 0; m < 4; ++m)
                #pragma unroll
                for (int n = 0; n < 2; ++n)
                    acc[m][n] = wmma4(afr[m], bfr[n], acc[m][n]);
        }
        __syncthreads();
        #pragma unroll
        for (int n = 0; n < 2; ++n) {
            const int col = (wv * 2 + n) * 16 + l15;
            #pragma unroll
            for (int m = 0; m < 4; ++m)
                #pragma unroll
                for (int r = 0; r < 8; ++r)
                    T[m * 16 + r + kh * 8][col] = acc[m][n][r];
        }
    }
    __syncthreads();

    // ---- phase 4: F = O * W2^T + c2  (K=256) -> X region ----
    {
        v8f acc[4][2];
        #pragma unroll
        for (int m = 0; m < 4; ++m)
            #pragma unroll
            for (int n = 0; n < 2; ++n) acc[m][n] = vz;
        for (int k = 0; k < DM; k += 4) {
            v2f afr[4], bfr[2];
            #pragma unroll
            for (int m = 0; m < 4; ++m)
                afr[m] = *(const v2f*)&T[m * 16 + l15][k + 2 * kh];
            #pragma unroll
            for (int n = 0; n < 2; ++n) {
                const int o = (wv * 2 + n) * 16 + l15;
                bfr[n] = *(const v2f*)&W2p[(size_t)((k >> 1) + kh) * 512 + o * 2];
            }
            #pragma unroll
            for (int m = 0; m < 4; ++m)
                #pragma unroll
                for (int n = 0; n < 2; ++n)
                    acc[m][n] = wmma4(afr[m], bfr[n], acc[m][n]);
        }
        #pragma unroll
        for (int n = 0; n < 2; ++n) {
            const int o = (wv * 2 + n) * 16 + l15;
            const float bo = c2[o];
            #pragma unroll
            for (int m = 0; m < 4; ++m)
                #pragma unroll
                for (int r = 0; r < 8; ++r)
                    X[m * 16 + r + kh * 8][o] = acc[m][n][r] + bo;
        }
    }
    __syncthreads();

    // ---- write out[b][o][hc*64+q][wc*64+p], coalesced 256B runs ----
    float* ob = out + (size_t)b * DM * HW + (size_t)(hc * 64 + q) * 256 + wc * 64;
    for (int i = t; i < 64 * DM; i += 256) {
        int o = i >> 6, p = i & 63;
        ob[(size_t)o * HW + p] = X[p][o];
    }
}

// ---------------------------------------------------------------------------
extern "C" void kernel_launch(void* const* d_in, const int* in_sizes, int n_in,
                              void* d_out, int out_size, void* d_ws, size_t ws_size,
                              hipStream_t stream) {
    const float* pf  = (const float*)d_in[0];
    const float* cw  = (const float*)d_in[1];
    const float* cb  = (const float*)d_in[2];
    const float* wq  = (const float*)d_in[3];
    const float* wqb = (const float*)d_in[4];
    const float* wk  = (const float*)d_in[5];
    const float* wkb = (const float*)d_in[6];
    const float* wvw = (const float*)d_in[7];
    const float* wvb = (const float*)d_in[8];
    const float* fc  = (const float*)d_in[9];
    const float* fcb = (const float*)d_in[10];

    float* ws  = (float*)d_ws;
    float* x   = ws;                    // 2*16*4096*256 = 33,554,432 floats
    float* Mp  = ws + 33554432;         // 65536
    float* W2p = Mp + 65536;            // 65536
    float* cv  = W2p + 65536;           // 256
    float* c2  = cv + 256;              // 256
    float* sc  = c2 + 256;              // 1

    precompute_mats<<<dim3(256, 2), 256, 0, stream>>>(wq, wk, wvw, fc, Mp, W2p);
    precompute_vecs<<<1, 256, 0, stream>>>(wk, wqb, wkb, fc, wvb, fcb, cv, c2, sc);
    conv_pe<<<2048, 256, 0, stream>>>(pf, cw, cb, x);
    attn_fused<<<2048, 256, 0, stream>>>(x, Mp, W2p, cv, c2, sc, (float*)d_out);
}